// SeasonalityBlock_12575664243332
// MI455X (gfx1250) — compile-verified
//
#include <hip/hip_runtime.h>
#include <math.h>

typedef __attribute__((ext_vector_type(2))) float v2f;
typedef __attribute__((ext_vector_type(8))) float v8f;

#define T_LEN   2048
#define NBINS   1024      // bins per half (Re cols 0..1023, Im cols 1024..2047)
#define NCOLS   (2 * NBINS)
#define NSEQ    2048      // 32 * 64
#define PRED    96
#define OUT_T   (T_LEN + PRED)   // 2144
#define KTOP    8
#define BATCH   32
#define DMODEL  64
#define TWO_PI_OVER_2048 3.0679615758e-03f

// ---------------------------------------------------------------------------
// Kernel 1: basis B[t][j], j<1024: cos(2*pi*j*t/2048), j>=1024: sin(...)
// Exact integer phase reduction mod 2048 -> small-argument sin/cos only.
// ---------------------------------------------------------------------------
__global__ void build_basis(float* __restrict__ Bm) {
    int idx = blockIdx.x * blockDim.x + threadIdx.x;   // idx = t*2048 + j
    if (idx >= T_LEN * NCOLS) return;
    int j = idx & 2047;
    int t = idx >> 11;
    int m = j & 1023;
    int p = (m * t) & 2047;          // exact phase index
    float th = (float)p * TWO_PI_OVER_2048;
    float sn, cs;
    __sincosf(th, &sn, &cs);
    Bm[idx] = (j < NBINS) ? cs : sn;
}

// ---------------------------------------------------------------------------
// Kernel 2: DFT GEMM via V_WMMA_F32_16X16X4_F32 with 4x2 register blocking.
// Each wave computes a 64(M) x 32(N) tile of C = 8 independent 16x16
// accumulators; per K-step of 4 it loads 4 A-frags + 2 B-frags (12 dwords)
// feeding 8 WMMAs (1.5 dw/WMMA instead of 4.0 unblocked).
// A 64-row M tile lies inside one batch slab (64 == DMODEL), so the wave
// shares a single X base pointer and all fragment loads are 64B-contiguous
// per 16-lane half.
// Fragment layout (32-bit A 16x4): lanes 0-15 hold M=0..15 K={0,1},
// lanes 16-31 hold M=0..15 K={2,3}; B mirrors with N across lanes.
// ---------------------------------------------------------------------------
#define MT 4   // M subtiles per wave (64 rows)
#define NT 2   // N subtiles per wave (32 cols)

__global__ void dft_wmma(const float* __restrict__ X,
                         const float* __restrict__ Bm,
                         float* __restrict__ C) {
    int wid  = (blockIdx.x << 3) | (threadIdx.x >> 5);   // 8 waves / block
    int lane = threadIdx.x & 31;
    int tm = wid >> 6;            // 0..31  (M tile of 64)
    int tn = wid & 63;            // 0..63  (N tile of 32)
    int m0 = tm << 6;
    int n0 = tn << 5;
    int li = lane & 15;
    int hf = lane >> 4;           // 0: K={0,1}, 1: K={2,3}

    const float* Xb = X + (size_t)(m0 >> 6) * (T_LEN * DMODEL); // batch slab
    const float* Bp = Bm + n0;

    v8f acc[MT][NT];
    #pragma unroll
    for (int i = 0; i < MT; ++i)
        #pragma unroll
        for (int j = 0; j < NT; ++j)
            acc[i][j] = (v8f){};

    #pragma unroll 2
    for (int k = 0; k < T_LEN; k += 4) {
        int k0 = k + (hf << 1);
        v2f av[MT], bv[NT];
        #pragma unroll
        for (int i = 0; i < MT; ++i) {
            av[i].x = Xb[(size_t)k0 * DMODEL + (i << 4) + li];
            av[i].y = Xb[(size_t)(k0 + 1) * DMODEL + (i << 4) + li];
        }
        #pragma unroll
        for (int j = 0; j < NT; ++j) {
            bv[j].x = Bp[(size_t)k0 * NCOLS + (j << 4) + li];
            bv[j].y = Bp[(size_t)(k0 + 1) * NCOLS + (j << 4) + li];
        }
        #pragma unroll
        for (int i = 0; i < MT; ++i)
            #pragma unroll
            for (int j = 0; j < NT; ++j)
                acc[i][j] = __builtin_amdgcn_wmma_f32_16x16x4_f32(
                    /*neg_a=*/false, av[i], /*neg_b=*/false, bv[j],
                    /*c_mod=*/(short)0, acc[i][j],
                    /*reuse_a=*/false, /*reuse_b=*/false);
    }

    // C/D layout: lanes 0-15: N=li, VGPR r -> M=r ; lanes 16-31: N=li, M=8+r
    #pragma unroll
    for (int i = 0; i < MT; ++i) {
        #pragma unroll
        for (int j = 0; j < NT; ++j) {
            float* cptr = C + (size_t)(m0 + (i << 4) + (hf << 3)) * NCOLS
                            + n0 + (j << 4) + li;
            #pragma unroll
            for (int r = 0; r < 8; ++r)
                cptr[(size_t)r * NCOLS] = acc[i][j][r];
        }
    }
}

// ---------------------------------------------------------------------------
// Kernel 3: per-sequence top-8 over mag^2 = Re^2 + Im^2, bins m=1..1023.
// One 256-thread block per sequence; 8 argmax rounds with LDS reduction,
// lowest-index preference on ties.
// ---------------------------------------------------------------------------
__global__ void topk_kernel(const float* __restrict__ C,
                            float* __restrict__ tre,
                            float* __restrict__ tim,
                            int*   __restrict__ tmi) {
    int s   = blockIdx.x;
    int tid = threadIdx.x;
    const float* row = C + (size_t)s * NCOLS;

    float mag[4]; int mm[4];
    #pragma unroll
    for (int q = 0; q < 4; ++q) {
        int m = tid + (q << 8);                 // 0..1023
        float rr = row[m];
        float ii = row[m + NBINS];
        float v  = rr * rr + ii * ii;
        if (m == 0) v = -1.0f;                  // bin 0 excluded
        mag[q] = v; mm[q] = m;
    }

    __shared__ float sv[256];
    __shared__ int   si[256];
    __shared__ int   chosen;

    for (int r = 0; r < KTOP; ++r) {
        float bv = -2.0f; int bi = 1 << 30;
        #pragma unroll
        for (int q = 0; q < 4; ++q) {
            if (mag[q] > bv || (mag[q] == bv && mm[q] < bi)) { bv = mag[q]; bi = mm[q]; }
        }
        sv[tid] = bv; si[tid] = bi;
        __syncthreads();
        for (int step = 128; step > 0; step >>= 1) {
            if (tid < step) {
                float ov = sv[tid + step]; int oi = si[tid + step];
                if (ov > sv[tid] || (ov == sv[tid] && oi < si[tid])) {
                    sv[tid] = ov; si[tid] = oi;
                }
            }
            __syncthreads();
        }
        if (tid == 0) {
            int cm = si[0];
            chosen = cm;
            tre[s * KTOP + r] = row[cm];
            tim[s * KTOP + r] = row[cm + NBINS];
            tmi[s * KTOP + r] = cm;
        }
        __syncthreads();
        int cm = chosen;
        #pragma unroll
        for (int q = 0; q < 4; ++q)
            if (mm[q] == cm) mag[q] = -1.0f;
        __syncthreads();
    }
}

// ---------------------------------------------------------------------------
// Kernel 4: out[b,tau,d] = (2/2048) * sum_k Re_k*cos(th) + SinPart_k*sin(th),
// th = 2*pi*((m_k*tau) mod 2048)/2048 (exact integer reduction).
// ---------------------------------------------------------------------------
__global__ void reconstruct(const float* __restrict__ tre,
                            const float* __restrict__ tim,
                            const int*   __restrict__ tmi,
                            float* __restrict__ out) {
    int idx = blockIdx.x * blockDim.x + threadIdx.x;
    if (idx >= BATCH * OUT_T * DMODEL) return;
    int d    = idx & 63;
    int rest = idx >> 6;
    int tau  = rest % OUT_T;
    int b    = rest / OUT_T;
    int s    = b * DMODEL + d;

    float acc = 0.0f;
    #pragma unroll
    for (int r = 0; r < KTOP; ++r) {
        int   m  = tmi[s * KTOP + r];
        int   p  = (m * tau) & 2047;
        float th = (float)p * TWO_PI_OVER_2048;
        float sn, cs;
        __sincosf(th, &sn, &cs);
        acc = fmaf(tre[s * KTOP + r], cs, acc);
        acc = fmaf(tim[s * KTOP + r], sn, acc);
    }
    out[idx] = acc * (2.0f / (float)T_LEN);
}

// ---------------------------------------------------------------------------
// Launcher. Workspace layout (floats):
//   [0, 4194304)         basis B   (16 MiB)
//   [4194304, 8388608)   C = Re/Im (16 MiB)
//   [8388608, +16384)    top-8 Re
//   [.., +16384)         top-8 SinPart
//   [.., +16384)         top-8 bin index (int)
// Total ~33.8 MB -- fully resident in MI455X's 192 MB L2.
// ---------------------------------------------------------------------------
extern "C" void kernel_launch(void* const* d_in, const int* in_sizes, int n_in,
                              void* d_out, int out_size, void* d_ws, size_t ws_size,
                              hipStream_t stream) {
    const float* X = (const float*)d_in[0];
    float* out = (float*)d_out;
    float* ws  = (float*)d_ws;

    float* Bm  = ws;
    float* C   = ws + (size_t)T_LEN * NCOLS;
    float* tre = C  + (size_t)NSEQ * NCOLS;
    float* tim = tre + NSEQ * KTOP;
    int*   tmi = (int*)(tim + NSEQ * KTOP);

    {   // basis: 2048*2048 elements
        int n = T_LEN * NCOLS;
        build_basis<<<(n + 255) / 256, 256, 0, stream>>>(Bm);
    }
    {   // GEMM: 32x64 wave-tiles (64x32 C each), 8 waves per 256-thread block
        dft_wmma<<<(32 * 64) / 8, 256, 0, stream>>>(X, Bm, C);
    }
    {   // top-8 per sequence
        topk_kernel<<<NSEQ, 256, 0, stream>>>(C, tre, tim, tmi);
    }
    {   // reconstruction
        int n = BATCH * OUT_T * DMODEL;
        reconstruct<<<(n + 255) / 256, 256, 0, stream>>>(tre, tim, tmi, out);
    }
}